// SCF_GRUCell_47218870452490
// MI455X (gfx1250) — compile-verified
//
#include <hip/hip_runtime.h>
#include <hip/hip_bf16.h>
#include <math.h>

#define N_OTHERS 400000
#define HID 48
#define INP 96
#define SP0 8
#define SP1 8
#define NBINS 64
#define R0f 0.5f
#define R1f 40.0f
#define RSTEP ((R1f - R0f) / (float)SP0)      /* 4.9375 */
#define TWO_PI 6.2831853071795864769f
#define TSTEP (TWO_PI / (float)SP1)
#define LPAD 52                                /* pad 48 -> 52 to spread LDS banks */

typedef __attribute__((ext_vector_type(2))) float v2f;
typedef __attribute__((ext_vector_type(8))) float v8f;

__global__ __launch_bounds__(256) void zero_ws_kernel(float* ws, int n) {
  for (int i = blockIdx.x * blockDim.x + threadIdx.x; i < n; i += gridDim.x * blockDim.x)
    ws[i] = 0.0f;
}

// Stage 1: polar binning + per-block LDS histogram of 48-wide hidden rows.
__global__ __launch_bounds__(256) void bin_accum_kernel(
    const float* __restrict__ loc_agent,
    const float* __restrict__ loc_others,
    const int*   __restrict__ loc_other_index,
    const float* __restrict__ hiddens,
    float* __restrict__ sp_sum,   /* [64*48] in ws */
    float* __restrict__ cnt)      /* [64]    in ws */
{
  __shared__ float lsum[NBINS * LPAD];
  __shared__ float lcnt[NBINS];
  const int tid = threadIdx.x;
  for (int m = tid; m < NBINS * LPAD; m += blockDim.x) lsum[m] = 0.0f;
  if (tid < NBINS) lcnt[tid] = 0.0f;
  __syncthreads();

  const float ax = loc_agent[0];
  const float ay = loc_agent[1];
  const int stride = gridDim.x * blockDim.x;
  for (int i = blockIdx.x * blockDim.x + tid; i < N_OTHERS; i += stride) {
    float dx = loc_others[2 * i + 0] - ax;
    float dy = loc_others[2 * i + 1] - ay;
    float dist = sqrtf(dx * dx + dy * dy);
    if (dist >= R0f && dist <= R1f) {
      float safe = (dist > 0.0f) ? dist : 1.0f;
      float ca = fminf(1.0f, fmaxf(-1.0f, dx / safe));
      float ac = acosf(ca);
      float theta = (dy < 0.0f) ? (TWO_PI - ac) : ac;
      int u = (int)floorf((dist - R0f) / RSTEP);
      u = min(max(u, 0), SP0 - 1);
      int v = (int)floorf(theta / TSTEP);
      v = min(max(v, 0), SP1 - 1);
      int bin = u * SP1 + v;
      int idx = loc_other_index[i];
      const float4* hrow = (const float4*)(hiddens + (size_t)idx * HID);
      float* dstb = &lsum[bin * LPAD];
#pragma unroll
      for (int q = 0; q < HID / 4; ++q) {
        float4 hv = hrow[q];               // coalesced 16B per lane, contiguous rows per wave
        atomicAdd(&dstb[4 * q + 0], hv.x); // ds_add_f32
        atomicAdd(&dstb[4 * q + 1], hv.y);
        atomicAdd(&dstb[4 * q + 2], hv.z);
        atomicAdd(&dstb[4 * q + 3], hv.w);
      }
      atomicAdd(&lcnt[bin], 1.0f);
    }
  }
  __syncthreads();

  // one merge per block into global accumulators
  for (int m = tid; m < NBINS * HID; m += blockDim.x) {
    int bin = m / HID, j = m - bin * HID;
    float s = lsum[bin * LPAD + j];
    if (s != 0.0f) atomicAdd(&sp_sum[m], s);
  }
  if (tid < NBINS) {
    float c = lcnt[tid];
    if (c != 0.0f) atomicAdd(&cnt[tid], c);
  }
}

// Stage 2+3: average bins, FC via V_WMMA_F32_16X16X4_F32, GRU gates, output.
// 384 threads = 12 full waves. Wave w: row-tile t=w/4 (rows 16t..16t+15),
// K-slice p=w%4 (K in [768p, 768p+768)). EXEC is all-ones for the WMMA region.
__global__ __launch_bounds__(384) void fc_gru_kernel(
    const float* __restrict__ sp_sum, const float* __restrict__ cnt,
    const float* __restrict__ loc_agent, const float* __restrict__ feature_img,
    const float* __restrict__ f_vel, const float* __restrict__ hidden,
    const float* __restrict__ W_fc, const float* __restrict__ b_fc,
    const float* __restrict__ w_iz, const float* __restrict__ w_hz,
    const float* __restrict__ b_iz, const float* __restrict__ b_hz,
    const float* __restrict__ w_in, const float* __restrict__ w_hn,
    const float* __restrict__ b_in, const float* __restrict__ b_hn,
    float* __restrict__ out)
{
  __shared__ float sp[NBINS * HID];     // 3072 fp32
  __shared__ float fpart[3][4][16];     // per-(tile,kslice) partial fsp
  __shared__ float input_x[INP];
  __shared__ float gates[2];
  const int tid = threadIdx.x;

  // sp = where(cnt > 1, sp_sum / cnt, sp_sum)   (3072 = 8 iters * 384 threads, uniform)
  for (int m = tid; m < NBINS * HID; m += 384) {
    float c = cnt[m / HID];
    float s = sp_sum[m];
    sp[m] = (c > 1.0f) ? (s / c) : s;
  }
  __syncthreads();

  {
    const int wave = tid >> 5;
    const int lane = tid & 31;
    const int half = lane >> 4;   // 0: K=0,1   1: K=2,3   (A and B row striping)
    const int l    = lane & 15;   // M row within tile for A
    const int t    = wave >> 2;   // row tile 0..2
    const int p    = wave & 3;    // K slice 0..3
    const int kbase = p * 768;
    const float* Wrow = W_fc + (size_t)(t * 16 + l) * (NBINS * HID);

    v8f acc = {};
    for (int k0 = kbase; k0 < kbase + 768; k0 += 4) {
      int ka = k0 + (half ? 2 : 0);
      v2f a, b;
      a.x = Wrow[ka];      // A: 16x4 f32, lanes 0-15 K=0/1, lanes 16-31 K=2/3
      a.y = Wrow[ka + 1];
      b.x = sp[ka];        // B: 4x16 f32, same sp chunk replicated in all 16 cols
      b.y = sp[ka + 1];
      acc = __builtin_amdgcn_wmma_f32_16x16x4_f32(
          /*neg_a=*/false, a, /*neg_b=*/false, b,
          /*c_mod=*/(short)0, acc, /*reuse_a=*/false, /*reuse_b=*/false);
    }
    // D col 0: lane 0 holds M=0..7 in acc[0..7], lane 16 holds M=8..15
    if (l == 0) {
#pragma unroll
      for (int j = 0; j < 8; ++j)
        fpart[t][p][half * 8 + j] = acc[j];
    }
  }

  // feature_agent (32) and f_vel (16) into input_x[0..47]
  if (tid < 32) {
    int u0 = 40 - (int)loc_agent[1];   // H/2 - int(agent_y), H = 80
    int v0 = (int)loc_agent[0];
    input_x[tid] = feature_img[tid * 6400 + u0 * 80 + v0];
  } else if (tid < 48) {
    input_x[tid] = f_vel[tid - 32];
  }
  __syncthreads();

  // fsp = relu(sum of 4 K-slice partials + b_fc) -> input_x[48..95]
  if (tid < 48) {
    int t = tid >> 4, m = tid & 15;
    float s = fpart[t][0][m] + fpart[t][1][m] + fpart[t][2][m] + fpart[t][3][m] + b_fc[tid];
    input_x[48 + tid] = fmaxf(s, 0.0f);
  }
  __syncthreads();

  if (tid == 0) {
    float sz = 0.0f, sn = 0.0f;
    for (int i = 0; i < INP; ++i) {
      float x = input_x[i];
      sz += x * w_iz[i];
      sn += x * w_in[i];
    }
    float hz = 0.0f, hn = 0.0f;
    for (int i = 0; i < HID; ++i) {
      float h = hidden[i];
      hz += h * w_hz[i];
      hn += h * w_hn[i];
    }
    gates[0] = 1.0f / (1.0f + expf(-(sz + b_iz[0] + hz + b_hz[0])));  // zt
    gates[1] = tanhf(sn + b_in[0] + hn + b_hn[0]);                    // nt
    // rt is computed but unused in the reference -> omitted
  }
  __syncthreads();

  if (tid < HID) {
    float zt = gates[0], nt = gates[1];
    out[tid] = (1.0f - zt) * nt + zt * hidden[tid];
  }
}

extern "C" void kernel_launch(void* const* d_in, const int* in_sizes, int n_in,
                              void* d_out, int out_size, void* d_ws, size_t ws_size,
                              hipStream_t stream) {
  const float* loc_agent       = (const float*)d_in[0];
  const float* loc_others      = (const float*)d_in[1];
  const int*   loc_other_index = (const int*)  d_in[2];
  const float* feature_img     = (const float*)d_in[3];
  const float* f_vel           = (const float*)d_in[4];
  const float* hiddens         = (const float*)d_in[5];
  const float* hidden          = (const float*)d_in[6];
  const float* W_fc            = (const float*)d_in[7];
  const float* b_fc            = (const float*)d_in[8];
  /* d_in[9..12] = weight_ir, weight_hr, bias_ir, bias_hr (rt is dead in reference) */
  const float* w_iz            = (const float*)d_in[13];
  const float* w_hz            = (const float*)d_in[14];
  const float* b_iz            = (const float*)d_in[15];
  const float* b_hz            = (const float*)d_in[16];
  const float* w_in            = (const float*)d_in[17];
  const float* w_hn            = (const float*)d_in[18];
  const float* b_in            = (const float*)d_in[19];
  const float* b_hn            = (const float*)d_in[20];
  (void)in_sizes; (void)n_in; (void)out_size; (void)ws_size;

  float* ws     = (float*)d_ws;
  float* sp_sum = ws;           // 3072 floats
  float* cnt    = ws + 3072;    // 64 floats

  zero_ws_kernel<<<4, 256, 0, stream>>>(ws, 3136);
  bin_accum_kernel<<<512, 256, 0, stream>>>(loc_agent, loc_others, loc_other_index,
                                            hiddens, sp_sum, cnt);
  fc_gru_kernel<<<1, 384, 0, stream>>>(sp_sum, cnt, loc_agent, feature_img, f_vel,
                                       hidden, W_fc, b_fc,
                                       w_iz, w_hz, b_iz, b_hz,
                                       w_in, w_hn, b_in, b_hn,
                                       (float*)d_out);
}